// SAGEConv_14422500180562
// MI455X (gfx1250) — compile-verified
//
#include <hip/hip_runtime.h>

typedef float v2f __attribute__((ext_vector_type(2)));
typedef float v8f __attribute__((ext_vector_type(8)));

#define WT_STRIDE 68          // 64 + 4 pad -> conflict-free b64 LDS reads
#define TILE_M    16
#define WAVES_PB  8           // 256 threads = 8 wave32

__device__ __forceinline__ v8f wmma_f32x4(v2f a, v2f b, v8f c) {
  // D = A(16x4,f32) * B(4x16,f32) + C(16x16,f32)
  return __builtin_amdgcn_wmma_f32_16x16x4_f32(
      /*neg_a=*/false, a, /*neg_b=*/false, b,
      /*c_mod=*/(short)0, c, /*reuse_a=*/false, /*reuse_b=*/false);
}

__global__ void zero_f4_kernel(float4* __restrict__ p, long n4) {
  long i = (long)blockIdx.x * blockDim.x + threadIdx.x;
  if (i < n4) p[i] = make_float4(0.f, 0.f, 0.f, 0.f);
}

__global__ __launch_bounds__(256) void sage_edge_kernel(
    const float* __restrict__ h_src, const float* __restrict__ efeat,
    const int* __restrict__ src, const int* __restrict__ dst,
    const unsigned char* __restrict__ fwd,     // numpy bool = 1 byte
    const float* __restrict__ W_I, const float* __restrict__ b_I,
    const float* __restrict__ W_O, const float* __restrict__ b_O,
    const float* __restrict__ W_R, const float* __restrict__ b_R,
    float* __restrict__ n_out, float* __restrict__ e_out,
    int E, int numTiles)
{
  extern __shared__ float smem[];
  // layout: WT_I | WT_O | WT_R | biases(3x64) | per-wave scratch
  float* wtI  = smem;
  float* wtO  = smem + 64 * WT_STRIDE;
  float* wtR  = smem + 2 * 64 * WT_STRIDE;
  float* bias = smem + 3 * 64 * WT_STRIDE;          // 192 floats
  float* wmem = bias + 3 * 64;

  const int tid = threadIdx.x;

  // ---- block-cooperative: transpose weights into LDS (W is [K=64][N=64]) ----
  for (int idx = tid; idx < 64 * 64; idx += blockDim.x) {
    int k = idx >> 6, n = idx & 63;
    wtI[n * WT_STRIDE + k] = W_I[idx];
    wtO[n * WT_STRIDE + k] = W_O[idx];
    wtR[n * WT_STRIDE + k] = W_R[idx];
  }
  if (tid < 64) {
    bias[tid]       = b_I[tid];
    bias[64 + tid]  = b_O[tid];
    bias[128 + tid] = b_R[tid];
  }
  __syncthreads();

  const int wave = tid >> 5;
  const int lane = tid & 31;
  const int n15  = lane & 15;
  const int hi   = lane >> 4;

  float* ldsC = wmem + wave * (2 * TILE_M * WT_STRIDE + 32); // comp = h - e
  float* ldsE = ldsC + TILE_M * WT_STRIDE;                   // raw e
  int*   ldsDst = (int*)(ldsE + TILE_M * WT_STRIDE);         // 16 ints
  int*   ldsMsk = ldsDst + 16;                               // 16 ints

  for (int tile = blockIdx.x * WAVES_PB + wave; tile < numTiles;
       tile += gridDim.x * WAVES_PB) {
    const int tb = tile * TILE_M;

    // ---- stage cols 0..15 of the 16x64 comp/e tiles into LDS (coalesced b128) ----
    {
      const int q  = lane & 3;          // float4 column block
      const int r0 = lane >> 2;         // rows 0..7
      const int r1 = r0 + 8;            // rows 8..15
      int e0 = tb + r0; if (e0 >= E) e0 = E - 1;
      int e1 = tb + r1; if (e1 >= E) e1 = E - 1;
      const int s0 = src[e0], s1 = src[e1];
      const float4 h0 = *(const float4*)(h_src + (size_t)s0 * 64 + q * 4);
      const float4 h1 = *(const float4*)(h_src + (size_t)s1 * 64 + q * 4);
      const float4 g0 = *(const float4*)(efeat + (size_t)e0 * 64 + q * 4);
      const float4 g1 = *(const float4*)(efeat + (size_t)e1 * 64 + q * 4);
      *(float4*)(ldsE + r0 * WT_STRIDE + q * 4) = g0;
      *(float4*)(ldsE + r1 * WT_STRIDE + q * 4) = g1;
      *(float4*)(ldsC + r0 * WT_STRIDE + q * 4) =
          make_float4(h0.x - g0.x, h0.y - g0.y, h0.z - g0.z, h0.w - g0.w);
      *(float4*)(ldsC + r1 * WT_STRIDE + q * 4) =
          make_float4(h1.x - g1.x, h1.y - g1.y, h1.z - g1.z, h1.w - g1.w);
      if (lane < 16) {
        int er = tb + lane; if (er >= E) er = E - 1;
        ldsDst[lane] = dst[er];
        ldsMsk[lane] = (int)fwd[er];
      }
    }
    // DS ops are in-order within a wave; just stop the compiler reordering.
    asm volatile("" ::: "memory");

    // ---- A fragments: lane holds A[m][4k+2hi], A[m][4k+2hi+1] (ISA layout) ----
    v2f aC[16], aE[16];
#pragma unroll
    for (int k = 0; k < 16; ++k) {
      const int off = n15 * WT_STRIDE + 4 * k + 2 * hi;
      aC[k] = *(const v2f*)(ldsC + off);
      aE[k] = *(const v2f*)(ldsE + off);
    }
    int dgt[8], fmv[8];
#pragma unroll
    for (int v = 0; v < 8; ++v) {       // C/D layout: VGPR v -> row v + 8*hi
      const int m = v + 8 * hi;
      dgt[v] = ldsDst[m];
      fmv[v] = ldsMsk[m];
    }

#pragma unroll
    for (int nt = 0; nt < 4; ++nt) {
      const int ncol = nt * 16 + n15;
      const float bi = bias[ncol], bo = bias[64 + ncol], br = bias[128 + ncol];
      v8f accI = {bi, bi, bi, bi, bi, bi, bi, bi};
      v8f accO = {bo, bo, bo, bo, bo, bo, bo, bo};
      v8f accR = {br, br, br, br, br, br, br, br};
      const int bb = ncol * WT_STRIDE + 2 * hi;   // B frag: WT[n][4k+2hi ..]
#pragma unroll
      for (int k = 0; k < 16; ++k) {
        const v2f bI = *(const v2f*)(wtI + bb + 4 * k);
        const v2f bO = *(const v2f*)(wtO + bb + 4 * k);
        const v2f bR = *(const v2f*)(wtR + bb + 4 * k);
        accI = wmma_f32x4(aC[k], bI, accI);
        accO = wmma_f32x4(aC[k], bO, accO);
        accR = wmma_f32x4(aE[k], bR, accR);
      }
#pragma unroll
      for (int v = 0; v < 8; ++v) {
        const int m  = v + 8 * hi;
        const int er = tb + m;
        if (er < E) {
          const float sel = fmv[v] ? accI[v] : accO[v];
          atomicAdd(n_out + (size_t)dgt[v] * 64 + ncol, sel);   // segment_sum
          e_out[(size_t)er * 64 + ncol] = accR[v];              // e @ W_R + b_R
        }
      }
    }
  }
}

extern "C" void kernel_launch(void* const* d_in, const int* in_sizes, int n_in,
                              void* d_out, int out_size, void* d_ws, size_t ws_size,
                              hipStream_t stream) {
  const float* h_src = (const float*)d_in[0];
  const float* efeat = (const float*)d_in[1];
  const int*   src   = (const int*)d_in[2];
  const int*   dst   = (const int*)d_in[3];
  const unsigned char* fwd = (const unsigned char*)d_in[4];
  const float* W_I = (const float*)d_in[5];
  const float* b_I = (const float*)d_in[6];
  const float* W_O = (const float*)d_in[7];
  const float* b_O = (const float*)d_in[8];
  const float* W_R = (const float*)d_in[9];
  const float* b_R = (const float*)d_in[10];

  const int N = in_sizes[0] / 64;
  const int E = in_sizes[2];

  float* out   = (float*)d_out;
  float* n_out = out;                       // [N,64]
  float* e_out = out + (size_t)N * 64;      // [E,64]

  // zero the atomic-accumulated node output
  const long n4 = (long)N * 16;             // N*64 / 4
  zero_f4_kernel<<<(int)((n4 + 255) / 256), 256, 0, stream>>>((float4*)n_out, n4);

  const int numTiles = (E + TILE_M - 1) / TILE_M;
  int blocks = (numTiles + WAVES_PB - 1) / WAVES_PB;
  if (blocks > 1280) blocks = 1280;         // persistent waves; amortize LDS fill

  const size_t shmem =
      (size_t)(3 * 64 * WT_STRIDE + 3 * 64 + WAVES_PB * (2 * TILE_M * WT_STRIDE + 32)) *
      sizeof(float);                        // ~121 KB, fine for 320 KB/WGP
  hipFuncSetAttribute((const void*)sage_edge_kernel,
                      hipFuncAttributeMaxDynamicSharedMemorySize, (int)shmem);

  sage_edge_kernel<<<blocks, 256, shmem, stream>>>(
      h_src, efeat, src, dst, fwd, W_I, b_I, W_O, b_O, W_R, b_R,
      n_out, e_out, E, numTiles);
}